// NeuralExecutor_17420387352967
// MI455X (gfx1250) — compile-verified
//
#include <hip/hip_runtime.h>
#include <hip/hip_bf16.h>
#include <stdint.h>

// Problem constants (from reference)
constexpr int cB = 4, cN = 512, cND = 2, cH = 32, cE = 32;
constexpr int BN = cB * cN;          // 2048 rows
constexpr int KNE = 36;              // ND+H=34 padded to multiple of 4

typedef float v2f __attribute__((ext_vector_type(2)));
typedef float v8f __attribute__((ext_vector_type(8)));

__device__ __forceinline__ v8f wmma4(v2f a, v2f b, v8f c) {
  // V_WMMA_F32_16X16X4_F32 : D = A(16x4) * B(4x16) + C(16x16), all f32
  return __builtin_amdgcn_wmma_f32_16x16x4_f32(
      false, a, false, b, (short)0, c, false, false);
}

// ---- CDNA5 async global<->LDS helpers (ASYNCcnt-tracked, per-lane 16B) ----
__device__ __forceinline__ uint32_t lds_addr32(const void* p) {
  // low 32 bits of the flat LDS-aperture address == LDS byte offset
  return (uint32_t)(uintptr_t)p;
}
__device__ __forceinline__ void async_ld_b128(void* lds_dst, const void* gsrc) {
  asm volatile("global_load_async_to_lds_b128 %0, %1, off"
               :: "v"(lds_addr32(lds_dst)),
                  "v"((unsigned long long)(uintptr_t)gsrc)
               : "memory");
}
__device__ __forceinline__ void async_st_b128(void* gdst, const void* lds_src) {
  asm volatile("global_store_async_from_lds_b128 %0, %1, off"
               :: "v"((unsigned long long)(uintptr_t)gdst),
                  "v"(lds_addr32(lds_src))
               : "memory");
}
__device__ __forceinline__ void wait_async() {
  asm volatile("s_wait_asynccnt 0x0" ::: "memory");
}

// ---------------------------------------------------------------------------
// prep: fold W_ee/b_ee/W_m/W_p into g[32], c0[32], s, p0; zero tau accum.
// ---------------------------------------------------------------------------
__global__ void prep_kernel(const float* __restrict__ W_ee, const float* __restrict__ b_ee,
                            const float* __restrict__ W_m, const float* __restrict__ b_m,
                            const float* __restrict__ W_p, const float* __restrict__ b_p,
                            float* __restrict__ gvec, float* __restrict__ c0vec,
                            float* __restrict__ sp, float* __restrict__ hsum) {
  int k = threadIdx.x;  // 0..31
  float g = 0.0f, c0 = b_m[k], s = 0.0f, p0 = b_p[0];
  for (int tt = 0; tt < cE; ++tt) {
    float wmc = W_m[(2 * cE + tt) * cH + k];
    g  += W_ee[tt] * wmc;
    c0 += b_ee[tt] * wmc;
    float wpe = W_p[2 * cH + tt];
    s  += W_ee[tt] * wpe;
    p0 += b_ee[tt] * wpe;
  }
  gvec[k] = g;
  c0vec[k] = c0;
  if (k == 0) { sp[0] = s; sp[1] = p0; }
  for (int idx = k; idx < cB * cH; idx += 32) hsum[idx] = 0.0f;
}

// ---------------------------------------------------------------------------
// encode: z = [x,h]@W_ne + b_ne; U = z@W_m[0:32]; V = z@W_m[32:64]
// K-order permuted to [h(32) | x(2) | pad(2)] so h rows async-load as aligned
// 16B chunks. Weights pre-packed in LDS in WMMA-B lane layout so each B
// operand is one ds_load_b64. 8 waves/block, one 16-row tile per wave.
// ---------------------------------------------------------------------------
__global__ __launch_bounds__(256) void encode_kernel(
    const float* __restrict__ x, const float* __restrict__ h,
    const float* __restrict__ W_ne, const float* __restrict__ b_ne,
    const float* __restrict__ W_m,
    float* __restrict__ zg, float* __restrict__ Ug, float* __restrict__ Vg) {
  __shared__ __align__(16) float sWneP[(KNE / 4) * 64 * 2];  // 1152
  __shared__ __align__(16) float sWaP[(cE / 4) * 64 * 2];    // 1024
  __shared__ __align__(16) float sWbP[(cE / 4) * 64 * 2];    // 1024
  __shared__ __align__(16) float sA[8][16][KNE];
  __shared__ __align__(16) float sZ[8][16][cE];

  const int t = threadIdx.x;
  const int wave = t >> 5, lane = t & 31;
  const int half = lane >> 4, ln = lane & 15;
  const int row0 = (blockIdx.x * 8 + wave) * 16;

  // async-stage h rows into sA cols 0..31 (16 rows x 8 chunks of 16B)
  for (int c = lane; c < 16 * 8; c += 32) {
    int r = c >> 3, c4 = c & 7;
    async_ld_b128(&sA[wave][r][c4 * 4], h + (size_t)(row0 + r) * cH + c4 * 4);
  }
  // x cols (32..33) and zero pad (34..35)
  {
    int r = lane >> 1, d = lane & 1;
    sA[wave][r][cH + d] = x[(row0 + r) * cND + d];
    sA[wave][r][cH + cND + d] = 0.0f;
  }

  // pack W_ne into WMMA-B layout (K-permuted to match [h|x|pad])
  for (int s = t; s < (KNE / 4) * 64; s += 256) {
    int lane2 = s & 31, nt = (s >> 5) & 1, step = s >> 6;
    int k0 = step * 4 + 2 * (lane2 >> 4);
    int n0 = nt * 16 + (lane2 & 15);
    auto srcw = [&](int k) -> float {
      if (k < cH)       return W_ne[(k + cND) * cE + n0];  // h rows
      if (k < cH + cND) return W_ne[(k - cH) * cE + n0];   // x rows
      return 0.0f;                                          // pad
    };
    sWneP[s * 2 + 0] = srcw(k0);
    sWneP[s * 2 + 1] = srcw(k0 + 1);
  }
  // pack W_m rows 0..31 (Wa) and 32..63 (Wb)
  for (int s = t; s < (cE / 4) * 64; s += 256) {
    int lane2 = s & 31, nt = (s >> 5) & 1, step = s >> 6;
    int k0 = step * 4 + 2 * (lane2 >> 4);
    int n0 = nt * 16 + (lane2 & 15);
    sWaP[s * 2 + 0] = W_m[k0 * cH + n0];
    sWaP[s * 2 + 1] = W_m[(k0 + 1) * cH + n0];
    sWbP[s * 2 + 0] = W_m[(cE + k0) * cH + n0];
    sWbP[s * 2 + 1] = W_m[(cE + k0 + 1) * cH + n0];
  }
  wait_async();
  __syncthreads();

  // --- z = A @ Wne + b_ne ---
  const v2f* wneP = (const v2f*)sWneP;
  v8f zacc[2];
#pragma unroll
  for (int nt = 0; nt < 2; ++nt) {
    float bias = b_ne[nt * 16 + ln];
    v8f c = {bias, bias, bias, bias, bias, bias, bias, bias};
#pragma unroll
    for (int kk = 0; kk < KNE; kk += 4) {
      v2f a = *(const v2f*)&sA[wave][ln][kk + 2 * half];
      c = wmma4(a, wneP[(kk >> 2) * 64 + nt * 32 + lane], c);
    }
    zacc[nt] = c;
  }
#pragma unroll
  for (int nt = 0; nt < 2; ++nt)
#pragma unroll
    for (int r = 0; r < 8; ++r)
      sZ[wave][r + half * 8][nt * 16 + ln] = zacc[nt][r];
  __syncthreads();

  // async-store z tile LDS -> global (16B chunks)
  for (int c = lane; c < 16 * 8; c += 32) {
    int r = c >> 3, c4 = c & 7;
    async_st_b128(zg + (size_t)(row0 + r) * cE + c4 * 4, &sZ[wave][r][c4 * 4]);
  }

  // --- U = z @ Wa, V = z @ Wb ---
  const v2f* waP = (const v2f*)sWaP;
  const v2f* wbP = (const v2f*)sWbP;
  v8f uacc[2] = {}, vacc[2] = {};
#pragma unroll
  for (int kk = 0; kk < cE; kk += 4) {
    v2f a = *(const v2f*)&sZ[wave][ln][kk + 2 * half];
#pragma unroll
    for (int nt = 0; nt < 2; ++nt) {
      uacc[nt] = wmma4(a, waP[(kk >> 2) * 64 + nt * 32 + lane], uacc[nt]);
      vacc[nt] = wmma4(a, wbP[(kk >> 2) * 64 + nt * 32 + lane], vacc[nt]);
    }
  }
#pragma unroll
  for (int nt = 0; nt < 2; ++nt)
#pragma unroll
    for (int r = 0; r < 8; ++r) {
      int rr = r + half * 8;
      Ug[(row0 + rr) * cE + nt * 16 + ln] = uacc[nt][r];
      Vg[(row0 + rr) * cE + nt * 16 + ln] = vacc[nt][r];
    }
}

// ---------------------------------------------------------------------------
// message: agg[b,i,k] = max_{j in mask} relu(U[i,k]+V[j,k]+e_ij*g[k]+c0[k]).
// Whole V[b] slab (64KB) async-loaded into LDS once per block; inner loop is
// stride-1 ds_load + __shfl broadcasts of e_feat/mask.
// ---------------------------------------------------------------------------
__global__ __launch_bounds__(256) void message_kernel(
    const float* __restrict__ e_feat, const int* __restrict__ adj,
    const float* __restrict__ Ug, const float* __restrict__ Vg,
    const float* __restrict__ gvec, const float* __restrict__ c0vec,
    float* __restrict__ aggg) {
  __shared__ __align__(16) float sV[cN * cH];  // 64 KB
  const int wave = threadIdx.x >> 5, lane = threadIdx.x & 31;
  const int row = blockIdx.x * 8 + wave;       // b*N + i (block spans one b)
  const int b = row >> 9, i = row & (cN - 1);

  const float* Vb = Vg + (size_t)b * cN * cH;
  for (int c = threadIdx.x; c < (cN * cH) / 4; c += 256)
    async_ld_b128(&sV[c * 4], Vb + (size_t)c * 4);

  const float Uk = Ug[row * cH + lane];
  const float gk = gvec[lane], c0k = c0vec[lane];
  const float* efrow = e_feat + (size_t)row * cN;
  const int* adrow = adj + (size_t)row * cN;

  wait_async();
  __syncthreads();

  float acc = -1e9f;
  for (int j0 = 0; j0 < cN; j0 += 32) {
    if (j0 + 32 < cN) {
      __builtin_prefetch(efrow + j0 + 32 + lane, 0, 3);
      __builtin_prefetch(adrow + j0 + 32 + lane, 0, 3);
    }
    float ef = efrow[j0 + lane];
    int valid = ((adrow[j0 + lane] > 0) || (j0 + lane == i)) ? 1 : 0;
    for (int jj = 0; jj < 32; ++jj) {
      float efv = __shfl(ef, jj, 32);
      int vv = __shfl(valid, jj, 32);
      if (vv) {  // uniform across wave
        float m = Uk + sV[(j0 + jj) * cH + lane] + efv * gk + c0k;
        m = fmaxf(m, 0.0f);
        acc = fmaxf(acc, m);
      }
    }
  }
  aggg[row * cH + lane] = acc;
}

// ---------------------------------------------------------------------------
// update: new_h = relu([z,agg]@W_u+b_u) via WMMA (K=64); then new_x, a/c
// projections, and tau row-sum accumulation. [z|agg] rows async-staged.
// ---------------------------------------------------------------------------
__global__ __launch_bounds__(256) void update_kernel(
    const float* __restrict__ zg, const float* __restrict__ aggg,
    const float* __restrict__ W_u, const float* __restrict__ b_u,
    const float* __restrict__ W_dec, const float* __restrict__ b_dec,
    const float* __restrict__ W_p,
    float* __restrict__ out_newx, float* __restrict__ out_newh,
    float* __restrict__ av, float* __restrict__ cvv, float* __restrict__ hsum) {
  __shared__ __align__(16) float sWuP[(2 * cE / 4) * 64 * 2];  // 2048
  __shared__ __align__(16) float sA[8][16][2 * cE];
  __shared__ __align__(16) float sNH[8][16][cH];

  const int t = threadIdx.x;
  const int wave = t >> 5, lane = t & 31;
  const int half = lane >> 4, ln = lane & 15;
  const int row0 = (blockIdx.x * 8 + wave) * 16;

  // async-stage [z|agg] rows (16 rows x 16 chunks of 16B)
  for (int c = lane; c < 16 * 16; c += 32) {
    int r = c >> 4, c4 = c & 15;
    const float* src = (c4 < 8) ? (zg + (size_t)(row0 + r) * cE + c4 * 4)
                                : (aggg + (size_t)(row0 + r) * cH + (c4 - 8) * 4);
    async_ld_b128(&sA[wave][r][c4 * 4], src);
  }
  // pack W_u (64x32) into WMMA-B layout
  for (int s = t; s < (2 * cE / 4) * 64; s += 256) {
    int lane2 = s & 31, nt = (s >> 5) & 1, step = s >> 6;
    int k0 = step * 4 + 2 * (lane2 >> 4);
    int n0 = nt * 16 + (lane2 & 15);
    sWuP[s * 2 + 0] = W_u[k0 * cH + n0];
    sWuP[s * 2 + 1] = W_u[(k0 + 1) * cH + n0];
  }
  wait_async();
  __syncthreads();

  const v2f* wuP = (const v2f*)sWuP;
#pragma unroll
  for (int nt = 0; nt < 2; ++nt) {
    float bias = b_u[nt * 16 + ln];
    v8f c = {bias, bias, bias, bias, bias, bias, bias, bias};
#pragma unroll
    for (int kk = 0; kk < 2 * cE; kk += 4) {
      v2f a = *(const v2f*)&sA[wave][ln][kk + 2 * half];
      c = wmma4(a, wuP[(kk >> 2) * 64 + nt * 32 + lane], c);
    }
#pragma unroll
    for (int r = 0; r < 8; ++r) {
      int rr = r + half * 8;
      float v = fmaxf(c[r], 0.0f);
      sNH[wave][rr][nt * 16 + ln] = v;
      out_newh[(row0 + rr) * cH + nt * 16 + ln] = v;
    }
  }
  __syncthreads();

  {  // new_x = [z,new_h] @ W_dec + b_dec  (16 rows x ND=2 -> 32 lane-tasks)
    int r = lane >> 1, d = lane & 1;
    float s = b_dec[d];
    for (int k2 = 0; k2 < cE; ++k2) s += sA[wave][r][k2] * W_dec[k2 * cND + d];
    for (int k2 = 0; k2 < cH; ++k2) s += sNH[wave][r][k2] * W_dec[(cE + k2) * cND + d];
    out_newx[(row0 + r) * cND + d] = s;
  }
  {  // a = new_h . W_p[0:32],  c = new_h . W_p[32:64]
    int r = lane >> 1, which = lane & 1;
    float s = 0.0f;
    const float* wp = W_p + which * cH;
    for (int k2 = 0; k2 < cH; ++k2) s += sNH[wave][r][k2] * wp[k2];
    if (which == 0) av[row0 + r] = s;
    else            cvv[row0 + r] = s;
  }
  {  // tau accumulator: sum over the 16 rows (tile lies within one batch)
    const int b = row0 >> 9;
    float hs = 0.0f;
    for (int r = 0; r < 16; ++r) hs += sNH[wave][r][lane];
    atomicAdd(&hsum[b * cH + lane], hs);
  }
}

// ---------------------------------------------------------------------------
// tau[b] = (hsum[b]/N) . W_t + b_t
// ---------------------------------------------------------------------------
__global__ void tau_kernel(const float* __restrict__ hsum, const float* __restrict__ W_t,
                           const float* __restrict__ b_t, float* __restrict__ out_tau) {
  int t = threadIdx.x;  // 128 threads: wave b, lane k
  int b = t >> 5, k = t & 31;
  float v = hsum[b * cH + k] * W_t[k];
#pragma unroll
  for (int off = 16; off > 0; off >>= 1) v += __shfl_down(v, off, 32);
  if (k == 0) out_tau[b] = v * (1.0f / cN) + b_t[0];
}

// ---------------------------------------------------------------------------
// pair: p[b,i,j] = mask ? a[b,i] + c[b,j] + s*e_feat + p0 : -inf
// ---------------------------------------------------------------------------
__global__ __launch_bounds__(256) void pair_kernel(
    const float* __restrict__ e_feat, const int* __restrict__ adj,
    const float* __restrict__ av, const float* __restrict__ cvv,
    const float* __restrict__ sp, float* __restrict__ out_p) {
  int idx = blockIdx.x * 256 + threadIdx.x;  // 0 .. B*N*N-1
  int j = idx & (cN - 1);
  int i = (idx >> 9) & (cN - 1);
  int b = idx >> 18;
  bool m = (adj[idx] > 0) || (i == j);
  float s = sp[0], p0 = sp[1];
  float val = av[b * cN + i] + cvv[b * cN + j] + s * e_feat[idx] + p0;
  out_p[idx] = m ? val : -__builtin_inff();
}

// ---------------------------------------------------------------------------
extern "C" void kernel_launch(void* const* d_in, const int* in_sizes, int n_in,
                              void* d_out, int out_size, void* d_ws, size_t ws_size,
                              hipStream_t stream) {
  (void)in_sizes; (void)n_in; (void)out_size; (void)ws_size;
  const float* x     = (const float*)d_in[0];
  const float* h     = (const float*)d_in[1];
  const int*   adj   = (const int*)d_in[2];
  const float* e_ft  = (const float*)d_in[3];
  const float* W_ne  = (const float*)d_in[4];
  const float* b_ne  = (const float*)d_in[5];
  const float* W_ee  = (const float*)d_in[6];
  const float* b_ee  = (const float*)d_in[7];
  const float* W_m   = (const float*)d_in[8];
  const float* b_m   = (const float*)d_in[9];
  const float* W_u   = (const float*)d_in[10];
  const float* b_u   = (const float*)d_in[11];
  const float* W_dec = (const float*)d_in[12];
  const float* b_dec = (const float*)d_in[13];
  const float* W_p   = (const float*)d_in[14];
  const float* b_p   = (const float*)d_in[15];
  const float* W_t   = (const float*)d_in[16];
  const float* b_t   = (const float*)d_in[17];

  float* ws    = (float*)d_ws;
  float* zg    = ws;               // 65536
  float* Ug    = ws + 65536;       // 65536
  float* Vg    = ws + 131072;      // 65536
  float* aggg  = ws + 196608;      // 65536
  float* gvec  = ws + 262144;      // 32
  float* c0vec = ws + 262176;      // 32
  float* sp    = ws + 262208;      // 2 (+pad)
  float* av    = ws + 262240;      // 2048
  float* cvv   = ws + 264288;      // 2048
  float* hsum  = ws + 266336;      // 128

  float* out      = (float*)d_out;
  float* out_newx = out;                        // B*N*ND = 4096
  float* out_p    = out + 4096;                 // B*N*N  = 1048576
  float* out_tau  = out + 4096 + 1048576;       // B      = 4
  float* out_newh = out + 4096 + 1048576 + 4;   // B*N*H  = 65536

  prep_kernel<<<1, 32, 0, stream>>>(W_ee, b_ee, W_m, b_m, W_p, b_p, gvec, c0vec, sp, hsum);
  encode_kernel<<<BN / 128, 256, 0, stream>>>(x, h, W_ne, b_ne, W_m, zg, Ug, Vg);
  message_kernel<<<BN / 8, 256, 0, stream>>>(e_ft, adj, Ug, Vg, gvec, c0vec, aggg);
  update_kernel<<<BN / 128, 256, 0, stream>>>(zg, aggg, W_u, b_u, W_dec, b_dec, W_p,
                                              out_newx, out_newh, av, cvv, hsum);
  tau_kernel<<<1, 128, 0, stream>>>(hsum, W_t, b_t, out_tau);
  pair_kernel<<<(cB * cN * cN) / 256, 256, 0, stream>>>(e_ft, adj, av, cvv, sp, out_p);
}